// MPSMetricLearnerEmbedCenter_68040871903478
// MI455X (gfx1250) — compile-verified
//
#include <hip/hip_runtime.h>
#include <math.h>

// ---------------------------------------------------------------------------
// MPS metric learner on MI455X (gfx1250, wave32, WMMA + Tensor Data Mover).
// fp32 throughout via V_WMMA_F32_16X16X4_F32 (precision-safe for the
// 1022-step multiplicative chain). Panels are staged LDS-side by the TDM
// (tensor_load_to_lds) with descriptor-padding providing the bank-conflict
// row pitch; one wave issues, all waves sync on s_wait_tensorcnt + barrier.
//   1) mps_sweep_kernel : grid (B/64, 2) -> left sweep (y=0), right sweep (y=1)
//   2) mps_final_kernel : out = (left (x) right) @ center, L2-normalized rows
// ---------------------------------------------------------------------------

typedef float v2f  __attribute__((ext_vector_type(2)));
typedef float v8f  __attribute__((ext_vector_type(8)));
typedef unsigned int uint32x4 __attribute__((ext_vector_type(4)));
typedef int   int32x4 __attribute__((ext_vector_type(4)));
typedef int   int32x8 __attribute__((ext_vector_type(8)));

#define WMMA_F32X4(a, b, c) \
  __builtin_amdgcn_wmma_f32_16x16x4_f32(false, (a), false, (b), (short)0, (c), false, false)

#if __has_builtin(__builtin_amdgcn_tensor_load_to_lds) && \
    __has_builtin(__builtin_amdgcn_s_wait_tensorcnt)
#define HAVE_TDM 1
#else
#define HAVE_TDM 0
#endif

constexpr int kL     = 1024;
constexpr int kD     = 64;
constexpr int kO     = 128;
constexpr int kB     = 4096;
constexpr int kSteps = 511;         // steps per sweep
constexpr int ROWS   = 64;          // batch rows per workgroup (4 waves x 16)
constexpr int NW     = 4;           // waves per workgroup
constexpr int NT     = 128;         // threads per workgroup

constexpr int PPADL  = 72;          // left panel pitch  (128 rows x 64 cols)
constexpr int PPADR  = 132;         // right/center pitch (64 rows x 128 cols)
constexpr int PANEL  = 128 * PPADL; // 9216 floats, also covers 64*132 = 8448
constexpr int EPAD   = 17;          // env tile pitch: env[col][row16]
constexpr int ENV    = kD * EPAD;   // 1088 floats per env buffer

// TDM pad encodings: interval = 2^(enc+1) DWORDs, amount = enc+1 DWORDs.
constexpr unsigned PADIV_L = 5;     // every 64 DWORDs
constexpr unsigned PADAM_L = 7;     // +8 DWORDs  -> pitch 72
constexpr unsigned PADIV_R = 6;     // every 128 DWORDs
constexpr unsigned PADAM_R = 3;     // +4 DWORDs  -> pitch 132

__device__ __forceinline__ void phi_of(float xv, float& c, float& s) {
  __sincosf(1.5707963267948966f * xv, &s, &c);
}

#if HAVE_TDM
// Issue a 1-D TDM copy of n_dw DWORDs from global to LDS with row padding.
// D# group0: count=1, lds_addr, 57-bit global_addr, type=2 ("image").
// D# group1: data_size=4B, pad_enable, pad_interval/amount, dims/tile/stride.
__device__ __forceinline__ void tdm_load(const void* lds_ptr, const void* gptr,
                                         unsigned n_dw, unsigned pad_iv,
                                         unsigned pad_am) {
  const unsigned lds_off = (unsigned)(uintptr_t)lds_ptr;   // low 32b = LDS byte offset
  const unsigned long long ga = (unsigned long long)(uintptr_t)gptr;
  uint32x4 g0;
  g0.x = 1u;                                   // count=1, no gather, user mode
  g0.y = lds_off;                              // lds_addr  (bits 63:32)
  g0.z = (unsigned)ga;                         // global_addr low
  g0.w = (unsigned)(ga >> 32) | (2u << 30);    // global_addr hi | type=2
  int32x8 g1;
  g1[0] = (int)((2u << 16) | (1u << 20) | (pad_iv << 22) | (pad_am << 25));
  g1[1] = (int)((n_dw & 0xFFFFu) << 16);       // tensor_dim0 lo16 (bits 79:48)
  g1[2] = (int)((n_dw >> 16) | (1u << 16));    // tensor_dim0 hi16 | tensor_dim1=1
  g1[3] = (int)((n_dw & 0xFFFFu) << 16);       // tile_dim0 (bits 127:112)
  g1[4] = 1;                                   // tile_dim1=1, tile_dim2=0
  g1[5] = (int)n_dw;                           // tensor_dim0_stride low
  g1[6] = 0;
  g1[7] = 0;
  int32x4 z4 = (int32x4)0;                     // groups 2/3 unused (<=2D)
#if defined(__clang_major__) && (__clang_major__ >= 23)
  int32x8 z8 = (int32x8)0;
  __builtin_amdgcn_tensor_load_to_lds(g0, g1, z4, z4, z8, 0);
#else
  __builtin_amdgcn_tensor_load_to_lds(g0, g1, z4, z4, 0);
#endif
}
#endif

// ---------------------------------------------------------------------------
// Sweep kernel: one workgroup = 64 batch rows, blockIdx.y selects sweep dir.
// Left sweep : D(batch16 x r64)  = U(env(x)phi) x panel[k=2l+p][r]
// Right sweep: D(l64 x batch16)  = panel[l][k=p*64+r] x U  (no LDS transpose)
// Per-step L2 normalization via shfl_xor reductions on the D-fragment layout.
// ---------------------------------------------------------------------------
__global__ __launch_bounds__(NT) void
mps_sweep_kernel(const float* __restrict__ x,
                 const float* __restrict__ left0,
                 const float* __restrict__ left_rest,
                 const float* __restrict__ right_rest,
                 const float* __restrict__ right_last,
                 float* __restrict__ lenv_out,
                 float* __restrict__ renv_out)
{
  extern __shared__ float smem[];
  float* panel0  = smem;
  float* panel1  = smem + PANEL;
  float* envbase = smem + 2 * PANEL;

  const int  tid     = threadIdx.x;
  const int  lane    = tid & 31;
  const int  wave    = tid >> 5;
  const int  m       = lane & 15;     // row/col index within 16-wide fragment
  const int  half    = lane >> 4;     // which half-wave (K-slot pair selector)
  const bool is_left = (blockIdx.y == 0);
  const int  b0      = blockIdx.x * ROWS;
  const int  row     = b0 + wave * 16 + m;   // this lane's batch row

  const unsigned pad_iv = is_left ? PADIV_L : PADIV_R;
  const unsigned pad_am = is_left ? PADAM_L : PADAM_R;

  float* envw = envbase + wave * (2 * ENV);  // per-wave ping-pong env tiles

  // step s reads panel from this source (511-step L2-resident streams)
  auto panel_src = [&](int s) -> const float* {
    return is_left ? left_rest  + (size_t)s * 8192
                   : right_rest + (size_t)(kSteps - 1 - s) * 8192;
  };

#if !HAVE_TDM
  auto stage = [&](float* pan, int s) {
    const float* src = panel_src(s);
    if (is_left) {
#pragma unroll
      for (int it = 0; it < 16; ++it) {
        const int i = tid * 4 + it * (NT * 4);
        float4 v = *(const float4*)(src + i);
        *(float4*)(pan + (i >> 6) * PPADL + (i & 63)) = v;
      }
    } else {
#pragma unroll
      for (int it = 0; it < 16; ++it) {
        const int i = tid * 4 + it * (NT * 4);
        float4 v = *(const float4*)(src + i);
        *(float4*)(pan + (i >> 7) * PPADR + (i & 127)) = v;
      }
    }
  };
#endif

  // ---- init env from the boundary tensor --------------------------------
  {
    const int site = is_left ? 0 : (kL - 1);
    float cph, sph;
    phi_of(x[(size_t)row * kL + site], cph, sph);
    const float* bt = is_left ? left0 : right_last;
    float vals[32];
    float ssl = 0.f;
#pragma unroll
    for (int cc = 0; cc < 32; ++cc) {
      const int col = half * 32 + cc;
      // left0 layout (p, r): bt[p*64 + r]; right_last layout (l, p): bt[2l + p]
      const float v = is_left ? (cph * bt[col]     + sph * bt[64 + col])
                              : (cph * bt[2 * col] + sph * bt[2 * col + 1]);
      vals[cc] = v;
      ssl += v * v;
    }
    ssl += __shfl_xor(ssl, 16, 32);                 // combine the two col-halves
    const float sc = 1.f / (sqrtf(ssl) + 1e-8f);
#pragma unroll
    for (int cc = 0; cc < 32; ++cc) {
      const int col = half * 32 + cc;
      envw[col * EPAD + m] = vals[cc] * sc;
    }
  }

  float xnext = x[(size_t)row * kL + (is_left ? 1 : (kL - 2))];

  // ---- stage panel for step 0 --------------------------------------------
#if HAVE_TDM
  if (wave == 0) {
    tdm_load(panel0, panel_src(0), 8192, pad_iv, pad_am);
    __builtin_amdgcn_s_wait_tensorcnt(0);
  }
#else
  stage(panel0, 0);
#endif
  __syncthreads();

  int pb = 0;
  for (int s = 0; s < kSteps; ++s) {
    float* pcur = ((s & 1) == 0) ? panel0 : panel1;
    float* pnxt = ((s & 1) == 0) ? panel1 : panel0;

    // launch async staging of the next panel; overlaps this step's WMMA work
    if (s + 1 < kSteps) {
#if HAVE_TDM
      if (wave == 0) tdm_load(pnxt, panel_src(s + 1), 8192, pad_iv, pad_am);
#else
      stage(pnxt, s + 1);
#endif
    }
    if (s + 2 < kSteps)                               // warm L2 ahead of the TDM
      __builtin_prefetch(panel_src(s + 2) + tid * 64, 0, 3);

    const float xc = xnext;
    if (s + 1 < kSteps)
      xnext = x[(size_t)row * kL + (is_left ? (2 + s) : (kL - 3 - s))];
    float cph, sph;
    phi_of(xc, cph, sph);

    const float* eb = envw + pb * ENV;
    float*       en = envw + (pb ^ 1) * ENV;

    v8f acc[4] = {};
    if (is_left) {
      // u[m, k=2l+p] = env[m,l] * phi[m,p];  B = panel[k][r]
#pragma unroll
      for (int t = 0; t < 32; ++t) {
        const float e = eb[(2 * t + half) * EPAD + m];
        v2f a; a.x = e * cph; a.y = e * sph;          // p = 0, 1
        const int kr = 4 * t + 2 * half;
#pragma unroll
        for (int j = 0; j < 4; ++j) {
          v2f b;
          b.x = pcur[kr * PPADL + 16 * j + m];
          b.y = pcur[(kr + 1) * PPADL + 16 * j + m];
          acc[j] = WMMA_F32X4(a, b, acc[j]);
        }
      }
      // normalize rows (batch row = v + 8*half), reduce over 16 lanes of half
      float ss[8];
#pragma unroll
      for (int v = 0; v < 8; ++v) {
        float s2 = 0.f;
#pragma unroll
        for (int j = 0; j < 4; ++j) s2 += acc[j][v] * acc[j][v];
        ss[v] = s2;
      }
#pragma unroll
      for (int mk = 1; mk <= 8; mk <<= 1)
#pragma unroll
        for (int v = 0; v < 8; ++v) ss[v] += __shfl_xor(ss[v], mk, 32);
#pragma unroll
      for (int v = 0; v < 8; ++v) ss[v] = 1.f / (sqrtf(ss[v]) + 1e-8f);
#pragma unroll
      for (int j = 0; j < 4; ++j)
#pragma unroll
        for (int v = 0; v < 8; ++v)
          en[(16 * j + m) * EPAD + (v + 8 * half)] = acc[j][v] * ss[v];
    } else {
      // D(l x batch): A = panel rows l (b64 reads), B = u[k=p*64+r]
#pragma unroll
      for (int t = 0; t < 32; ++t) {
        const int   kr = 4 * t + 2 * half;
        const float ph = (t >> 4) ? sph : cph;        // p = k >> 6
        v2f bu;
        bu.x = eb[( kr      & 63) * EPAD + m] * ph;
        bu.y = eb[((kr + 1) & 63) * EPAD + m] * ph;
#pragma unroll
        for (int j = 0; j < 4; ++j) {
          v2f a = *(const v2f*)(pcur + (16 * j + m) * PPADR + kr);  // 8B aligned
          acc[j] = WMMA_F32X4(a, bu, acc[j]);
        }
      }
      // normalize per batch column m: partial(lane) + partial(lane^16)
      float s2 = 0.f;
#pragma unroll
      for (int j = 0; j < 4; ++j)
#pragma unroll
        for (int v = 0; v < 8; ++v) s2 += acc[j][v] * acc[j][v];
      s2 += __shfl_xor(s2, 16, 32);
      const float sc = 1.f / (sqrtf(s2) + 1e-8f);
#pragma unroll
      for (int j = 0; j < 4; ++j)
#pragma unroll
        for (int v = 0; v < 8; ++v)
          en[(16 * j + v + 8 * half) * EPAD + m] = acc[j][v] * sc;
    }

    pb ^= 1;
#if HAVE_TDM
    if (wave == 0 && s + 1 < kSteps) __builtin_amdgcn_s_wait_tensorcnt(0);
#endif
    __syncthreads();
  }

  // ---- write final env tile to workspace ---------------------------------
  const float* ef   = envw + pb * ENV;
  float*       outp = is_left ? lenv_out : renv_out;
#pragma unroll
  for (int cc = 0; cc < 32; ++cc) {
    const int col = half * 32 + cc;
    outp[(size_t)row * kD + col] = ef[col * EPAD + m];
  }
}

// ---------------------------------------------------------------------------
// Final contraction: out[b,o] = sum_{l,r} left[b,l] right[b,r] center[l,r,o]
// K = 4096 chunked by l (64 chunks): within a chunk left[b,l] is a scalar per
// lane, so A-frag = left*right, B-frag = center rows (TDM-staged, double buf).
// ---------------------------------------------------------------------------
__global__ __launch_bounds__(NT) void
mps_final_kernel(const float* __restrict__ center,
                 const float* __restrict__ lenv,
                 const float* __restrict__ renv,
                 float* __restrict__ out)
{
  extern __shared__ float smem[];
  float* cs0     = smem;                      // 64 x PPADR center chunk (x2)
  float* cs1     = smem + 64 * PPADR;
  float* envbase = smem + 2 * 64 * PPADR;

  const int tid  = threadIdx.x;
  const int lane = tid & 31;
  const int wave = tid >> 5;
  const int m    = lane & 15;
  const int half = lane >> 4;
  const int b0   = blockIdx.x * ROWS;
  const int row  = b0 + wave * 16 + m;

  float* lt = envbase + wave * (2 * ENV);
  float* rt = lt + ENV;

#pragma unroll
  for (int cc = 0; cc < 32; ++cc) {
    const int col = half * 32 + cc;
    lt[col * EPAD + m] = lenv[(size_t)row * kD + col];
    rt[col * EPAD + m] = renv[(size_t)row * kD + col];
  }

#if !HAVE_TDM
  auto stage = [&](float* cs, int c) {
    const float* src = center + (size_t)c * (64 * kO);
#pragma unroll
    for (int it = 0; it < 16; ++it) {
      const int i = tid * 4 + it * (NT * 4);
      float4 v = *(const float4*)(src + i);
      *(float4*)(cs + (i >> 7) * PPADR + (i & 127)) = v;
    }
  };
#endif

#if HAVE_TDM
  if (wave == 0) {
    tdm_load(cs0, center, 8192, PADIV_R, PADAM_R);
    __builtin_amdgcn_s_wait_tensorcnt(0);
  }
#else
  stage(cs0, 0);
#endif
  __syncthreads();

  v8f acc[8] = {};
  for (int c = 0; c < 64; ++c) {              // chunk index == l
    float* cs = ((c & 1) == 0) ? cs0 : cs1;
    float* cn = ((c & 1) == 0) ? cs1 : cs0;
    if (c + 1 < 64) {
#if HAVE_TDM
      if (wave == 0)
        tdm_load(cn, center + (size_t)(c + 1) * (64 * kO), 8192, PADIV_R, PADAM_R);
#else
      stage(cn, c + 1);
#endif
    }

    const float lv = lt[c * EPAD + m];        // left[b(m), l=c]
#pragma unroll
    for (int t = 0; t < 16; ++t) {
      const int kr = 4 * t + 2 * half;        // r index
      v2f a;
      a.x = lv * rt[kr * EPAD + m];
      a.y = lv * rt[(kr + 1) * EPAD + m];
#pragma unroll
      for (int j = 0; j < 8; ++j) {
        v2f b;
        b.x = cs[kr * PPADR + 16 * j + m];
        b.y = cs[(kr + 1) * PPADR + 16 * j + m];
        acc[j] = WMMA_F32X4(a, b, acc[j]);
      }
    }

#if HAVE_TDM
    if (wave == 0 && c + 1 < 64) __builtin_amdgcn_s_wait_tensorcnt(0);
#endif
    __syncthreads();
  }

  // L2-normalize output rows (128 cols) and store
  float ss[8];
#pragma unroll
  for (int v = 0; v < 8; ++v) {
    float s2 = 0.f;
#pragma unroll
    for (int j = 0; j < 8; ++j) s2 += acc[j][v] * acc[j][v];
    ss[v] = s2;
  }
#pragma unroll
  for (int mk = 1; mk <= 8; mk <<= 1)
#pragma unroll
    for (int v = 0; v < 8; ++v) ss[v] += __shfl_xor(ss[v], mk, 32);
#pragma unroll
  for (int v = 0; v < 8; ++v) ss[v] = 1.f / fmaxf(sqrtf(ss[v]), 1e-12f);

#pragma unroll
  for (int j = 0; j < 8; ++j)
#pragma unroll
    for (int v = 0; v < 8; ++v) {
      const int orow = b0 + wave * 16 + v + 8 * half;
      out[(size_t)orow * kO + 16 * j + m] = acc[j][v] * ss[v];
    }
}

// ---------------------------------------------------------------------------
extern "C" void kernel_launch(void* const* d_in, const int* in_sizes, int n_in,
                              void* d_out, int out_size, void* d_ws, size_t ws_size,
                              hipStream_t stream) {
  const float* x          = (const float*)d_in[0];
  const float* left0      = (const float*)d_in[1];
  const float* left_rest  = (const float*)d_in[2];
  const float* center     = (const float*)d_in[3];
  const float* right_rest = (const float*)d_in[4];
  const float* right_last = (const float*)d_in[5];
  float*       out        = (float*)d_out;

  float* ws   = (float*)d_ws;
  float* lenv = ws;                       // B x D
  float* renv = ws + (size_t)kB * kD;     // B x D

  const size_t smem_sweep = (size_t)(2 * PANEL + NW * 2 * ENV) * sizeof(float);
  const size_t smem_final = (size_t)(2 * 64 * PPADR + NW * 2 * ENV) * sizeof(float);

  dim3 gsweep(kB / ROWS, 2, 1);           // 64 batch tiles x {left,right}
  mps_sweep_kernel<<<gsweep, NT, smem_sweep, stream>>>(
      x, left0, left_rest, right_rest, right_last, lenv, renv);

  mps_final_kernel<<<kB / ROWS, NT, smem_final, stream>>>(center, lenv, renv, out);
}